// HadamardTransform_10419590660574
// MI455X (gfx1250) — compile-verified
//
#include <hip/hip_runtime.h>

// FWHT-1024 via CDNA5 fp32 WMMA: H1024 = H16 (x) H16 (x) H4.
// One wave32 = one row of 1024 floats, staged through a private padded LDS slab.
// Memory-bound problem (512MB traffic @ 23.3 TB/s ~= 22us); WMMA fp32 is exact
// for +/-1 Hadamard factors, so correctness matches the fp32 reference.

typedef __attribute__((ext_vector_type(2))) float v2f;
typedef __attribute__((ext_vector_type(8))) float v8f;

#define ROW_STRIDE 68  // 64 data floats + 4 pad per 'a' row (bank-conflict padding)

__global__ __launch_bounds__(256) void fwht1024_wmma_kernel(
    const float* __restrict__ x,
    const float* __restrict__ scale,
    const float* __restrict__ shift,
    float* __restrict__ out,
    int nrows)
{
    __shared__ float lds_all[8][16 * ROW_STRIDE];

    const int lane = threadIdx.x & 31;
    const int wave = threadIdx.x >> 5;
    const int row  = blockIdx.x * 8 + wave;
    if (row >= nrows) return;

    float* lds = lds_all[wave];

    const int hi = lane >> 4;   // 0/1: which half of the wave
    const int lo = lane & 15;   // lane within half

    // ---- Load one row (1024 floats) global -> LDS, 128-bit coalesced ----
    const float4* in4 = reinterpret_cast<const float4*>(x) + (size_t)row * 256;
    #pragma unroll
    for (int i = 0; i < 8; ++i) {
        float4 v = in4[i * 32 + lane];
        // flat element index = (lane + 32*i)*4 -> a = 2*i + hi, offset = 4*lo
        int a = 2 * i + hi;
        *reinterpret_cast<float4*>(&lds[a * ROW_STRIDE + 4 * lo]) = v;
    }

    // ---- H16 as WMMA A-operand, chunked K=4 (layout: m=lo, k=j+2*hi per chunk) ----
    // H16[m][k] = (-1)^popcount(m & k); H16 is symmetric (Sylvester), so the same
    // operand serves both contraction stages.
    v2f A[4];
    #pragma unroll
    for (int k0 = 0; k0 < 4; ++k0) {
        int ka = 4 * k0 + 2 * hi;
        A[k0].x = (__popc(lo & ka) & 1) ? -1.0f : 1.0f;
        A[k0].y = (__popc(lo & (ka + 1)) & 1) ? -1.0f : 1.0f;
    }

    // ---- Stage 1: contract stride-64 axis a'.  View row as X[a'][m64], m64 = t*16+n.
    // D[a][n] = sum_{a'} H16[a][a'] * X[a'][t*16+n]
    #pragma unroll
    for (int t = 0; t < 4; ++t) {
        v8f d = {};
        #pragma unroll
        for (int k0 = 0; k0 < 4; ++k0) {
            int ka = 4 * k0 + 2 * hi;                 // B layout: n=lo, k=j+2*hi
            v2f b;
            b.x = lds[ka * ROW_STRIDE + t * 16 + lo];
            b.y = lds[(ka + 1) * ROW_STRIDE + t * 16 + lo];
            d = __builtin_amdgcn_wmma_f32_16x16x4_f32(false, A[k0], false, b,
                                                      (short)0, d, false, false);
        }
        #pragma unroll
        for (int v = 0; v < 8; ++v) {                 // D layout: m=v+8*hi, n=lo
            int m = v + 8 * hi;
            lds[m * ROW_STRIDE + t * 16 + lo] = d[v];
        }
    }

    // ---- Stage 2: contract stride-4 axis b'.  For fixed c:
    // D[b][a] = sum_{b'} H16[b][b'] * Y[a][b'*4+c]   (a rides the N axis)
    #pragma unroll
    for (int c = 0; c < 4; ++c) {
        v8f d = {};
        #pragma unroll
        for (int k0 = 0; k0 < 4; ++k0) {
            int ka = 4 * k0 + 2 * hi;
            v2f b;
            b.x = lds[lo * ROW_STRIDE + ka * 4 + c];
            b.y = lds[lo * ROW_STRIDE + (ka + 1) * 4 + c];
            d = __builtin_amdgcn_wmma_f32_16x16x4_f32(false, A[k0], false, b,
                                                      (short)0, d, false, false);
        }
        #pragma unroll
        for (int v = 0; v < 8; ++v) {
            int m = v + 8 * hi;
            lds[lo * ROW_STRIDE + m * 4 + c] = d[v];
        }
    }

    // ---- Stage 3: H4 butterfly on c, fuse 1/sqrt(1024) + scale/shift, store ----
    const float4* sc4 = reinterpret_cast<const float4*>(scale);
    const float4* sh4 = reinterpret_cast<const float4*>(shift);
    float4* out4 = reinterpret_cast<float4*>(out) + (size_t)row * 256;
    const float inv = 0.03125f;  // 1/32 = 1/sqrt(1024), exact in fp32
    #pragma unroll
    for (int i = 0; i < 8; ++i) {
        int G = lane + 32 * i;          // float4 group index within the row
        int a = G >> 4, b = G & 15;
        float4 w = *reinterpret_cast<const float4*>(&lds[a * ROW_STRIDE + b * 4]);
        float p01 = w.x + w.y, m01 = w.x - w.y;
        float p23 = w.z + w.w, m23 = w.z - w.w;
        float y0 = p01 + p23;   // c=0: + + + +
        float y1 = m01 + m23;   // c=1: + - + -
        float y2 = p01 - p23;   // c=2: + + - -
        float y3 = m01 - m23;   // c=3: + - - +
        float4 s = sc4[G];
        float4 h = sh4[G];
        float4 r;
        r.x = fmaf(y0 * inv, s.x, h.x);
        r.y = fmaf(y1 * inv, s.y, h.y);
        r.z = fmaf(y2 * inv, s.z, h.z);
        r.w = fmaf(y3 * inv, s.w, h.w);
        out4[G] = r;
    }
}

extern "C" void kernel_launch(void* const* d_in, const int* in_sizes, int n_in,
                              void* d_out, int out_size, void* d_ws, size_t ws_size,
                              hipStream_t stream) {
    const float* x     = (const float*)d_in[0];
    const float* scale = (const float*)d_in[1];
    const float* shift = (const float*)d_in[2];
    float* out = (float*)d_out;

    int nrows  = in_sizes[0] / 1024;       // 65536 rows of 1024 fp32
    int blocks = (nrows + 7) / 8;          // 8 waves (rows) per 256-thread block
    fwht1024_wmma_kernel<<<blocks, 256, 0, stream>>>(x, scale, shift, out, nrows);
}